// EMRouting_9466107920832
// MI455X (gfx1250) — compile-verified
//
#include <hip/hip_runtime.h>
#include <math.h>

#define B_    32
#define I_    2048
#define O_    64
#define P_    16
#define CH_   16            // I-chunks per batch
#define IPC_  (I_/CH_)      // 128 rows per workgroup
#define COLS_ 1024          // O_*P_ floats per (b,i) row
#define SLOT_ 1152          // 64 o * (16 p + rsum + den)
#define EPS_  1e-9f

typedef __attribute__((ext_vector_type(2))) float v2f;
typedef __attribute__((ext_vector_type(8))) float v8f;

// ---- workspace layout (floats) ----
#define WS_PART 0
#define WS_MU   (CH_*B_*SLOT_)        // 589824
#define WS_A    (WS_MU + B_*O_*P_)    // +32768
#define WS_LOGA (WS_A + B_*O_)        // +2048
// total = 626688 floats (~2.5 MB)

// -------------------------------------------------------------------------
// Pass 0: column sums of votes over an I-chunk via V_WMMA_F32_16X16X4_F32.
// A = all-ones (16x4), B[k][n] = votes[i+k, col+n]  =>  every D row holds
// the running column sums in full f32. One WMMA per 4 rows x 16 cols.
// -------------------------------------------------------------------------
__global__ __launch_bounds__(256) void sum_votes_wmma(const float* __restrict__ votes,
                                                      float* __restrict__ part) {
  const int b    = blockIdx.x / CH_;
  const int ch   = blockIdx.x % CH_;
  const int tid  = threadIdx.x;
  const int wave = tid >> 5;
  const int lane = tid & 31;
  const int n    = lane & 15;
  const int kx   = (lane >> 4) * 2;          // B VGPR0: K0 (lanes 0-15) / K2 (16-31)
  const int i0   = ch * IPC_;
  const int colbase = wave * 128;            // this wave covers 8 col-groups of 16

  v2f ones; ones.x = 1.0f; ones.y = 1.0f;
  v8f acc[8];
  for (int q = 0; q < 8; ++q)
    for (int e = 0; e < 8; ++e) acc[q][e] = 0.0f;

  const size_t base_bi = ((size_t)b * I_ + (size_t)i0) * COLS_;
  for (int i = 0; i < IPC_; i += 4) {
    const size_t r0 = base_bi + (size_t)(i + kx)     * COLS_;
    const size_t r1 = base_bi + (size_t)(i + kx + 1) * COLS_;   // B VGPR1: K1 / K3
#pragma unroll
    for (int q = 0; q < 8; ++q) {
      const int col = colbase + q * 16 + n;
      v2f bv;
      bv.x = votes[r0 + col];
      bv.y = votes[r1 + col];
      acc[q] = __builtin_amdgcn_wmma_f32_16x16x4_f32(
          false, ones, false, bv, (short)0, acc[q], false, false);
    }
  }

  float* pb = part + ((size_t)ch * B_ + b) * SLOT_;
  if (lane < 16) {                           // D row M=0: VGPR0, lanes 0-15 = N
#pragma unroll
    for (int q = 0; q < 8; ++q) pb[colbase + q * 16 + lane] = acc[q][0];
  }
}

// -------------------------------------------------------------------------
// Init finalize: Mu = l2norm(sum), A = 1/O, logA = log(A+eps)
// -------------------------------------------------------------------------
__global__ __launch_bounds__(64) void init_finalize(const float* __restrict__ part,
                                                    float* __restrict__ mu,
                                                    float* __restrict__ Ah,
                                                    float* __restrict__ logA) {
  const int b = blockIdx.x, o = threadIdx.x;
  float m[16];
#pragma unroll
  for (int p = 0; p < 16; ++p) m[p] = 0.0f;
  for (int ch = 0; ch < CH_; ++ch) {
    const float* pb = part + ((size_t)ch * B_ + b) * SLOT_ + o * 16;
#pragma unroll
    for (int p = 0; p < 16; ++p) m[p] += pb[p];
  }
  float sq = 0.0f;
#pragma unroll
  for (int p = 0; p < 16; ++p) sq += m[p] * m[p];
  const float rn = rsqrtf(fmaxf(sq, 1e-12f));
#pragma unroll
  for (int p = 0; p < 16; ++p) mu[(size_t)b * 1024 + o * 16 + p] = m[p] * rn;
  Ah[b * 64 + o]   = 1.0f / (float)O_;
  logA[b * 64 + o] = logf(1.0f / (float)O_ + EPS_);
}

// -------------------------------------------------------------------------
// Routing pass: one wave streams 4KB rows; lane l owns (o = l/4 + 8j, p-chunk
// l%4). Quad shfl-reduces give per-o norms/dots; wave shfl-reduces give the
// softmax over 64 o. Deterministic register->LDS->global partial reduction.
// -------------------------------------------------------------------------
__global__ __launch_bounds__(256) void routing_pass(const float* __restrict__ votes,
                                                    const float* __restrict__ act,
                                                    const float* __restrict__ mu,
                                                    const float* __restrict__ Ah,
                                                    const float* __restrict__ logA,
                                                    float* __restrict__ part) {
  __shared__ float4 smu4[256];
  __shared__ float  sA[64], slA[64];
  __shared__ float  lacc[SLOT_];

  const int b    = blockIdx.x / CH_;
  const int ch   = blockIdx.x % CH_;
  const int tid  = threadIdx.x;
  const int wave = tid >> 5;
  const int lane = tid & 31;
  const int pc   = lane & 3;      // p-chunk
  const int oq   = lane >> 2;     // o low bits

  smu4[tid] = ((const float4*)mu)[(size_t)b * 256 + tid];
  if (tid < 64) { sA[tid] = Ah[b * 64 + tid]; slA[tid] = logA[b * 64 + tid]; }
  for (int idx = tid; idx < SLOT_; idx += 256) lacc[idx] = 0.0f;
  __syncthreads();

  float accn[8][4], accr[8], accd[8];
#pragma unroll
  for (int j = 0; j < 8; ++j) {
    accn[j][0] = accn[j][1] = accn[j][2] = accn[j][3] = 0.0f;
    accr[j] = 0.0f; accd[j] = 0.0f;
  }

  const int i0 = ch * IPC_;
  for (int t = 0; t < 16; ++t) {
    const int r = i0 + wave + 8 * t;
    const size_t bi = (size_t)b * I_ + r;
    const float4* vrow = ((const float4*)votes) + bi * 256;
    const float a_in = act[bi];

    float4 v[8];
#pragma unroll
    for (int j = 0; j < 8; ++j) v[j] = vrow[lane + 32 * j];

    if (t + 1 < 16)  // global_prefetch_b8 for this wave's next row (4KB via 32 lanes)
      __builtin_prefetch((const void*)(((const float4*)votes) + (bi + 8) * 256 + lane * 8), 0, 0);

    float logit[8], Rw[8];
    float lmax = -3.0e38f;
#pragma unroll
    for (int j = 0; j < 8; ++j) {
      const int o = oq + 8 * j;
      const float4 m4 = smu4[o * 4 + pc];
      const float4 vj = v[j];
      float sq = vj.x * vj.x + vj.y * vj.y + vj.z * vj.z + vj.w * vj.w;
      float dt = vj.x * m4.x + vj.y * m4.y + vj.z * m4.z + vj.w * m4.w;
      sq += __shfl_xor(sq, 1, 32); sq += __shfl_xor(sq, 2, 32);
      dt += __shfl_xor(dt, 1, 32); dt += __shfl_xor(dt, 2, 32);
      const float rn = rsqrtf(fmaxf(sq, 1e-12f));
      const float d  = dt * rn;                         // votes_norm . Mu
      const float cv = 1.0f - d * d;                    // BETA_V = 1
      const float ca = fabsf(a_in - sA[o]);             // BETA_A = 1
      const float lg = -(cv + ca) + slA[o];
      logit[j] = lg;
      lmax = fmaxf(lmax, lg);
    }
#pragma unroll
    for (int m = 16; m >= 1; m >>= 1) lmax = fmaxf(lmax, __shfl_xor(lmax, m, 32));
    float lsum = 0.0f;
#pragma unroll
    for (int j = 0; j < 8; ++j) { const float e = __expf(logit[j] - lmax); Rw[j] = e; lsum += e; }
#pragma unroll
    for (int m = 16; m >= 1; m >>= 1) lsum += __shfl_xor(lsum, m, 32);
    const float inv = 4.0f / lsum;                      // each o counted 4x in wave sum
#pragma unroll
    for (int j = 0; j < 8; ++j) {
      const float Rj = Rw[j] * inv;
      accn[j][0] += Rj * v[j].x; accn[j][1] += Rj * v[j].y;
      accn[j][2] += Rj * v[j].z; accn[j][3] += Rj * v[j].w;
      accr[j] += Rj * a_in;
      accd[j] += Rj;
    }
  }

  // deterministic cross-wave reduce (fixed wave order)
  for (int w = 0; w < 8; ++w) {
    if (wave == w) {
#pragma unroll
      for (int j = 0; j < 8; ++j) {
        const int o = oq + 8 * j;
        const int base = o * 18 + pc * 4;
        lacc[base + 0] += accn[j][0]; lacc[base + 1] += accn[j][1];
        lacc[base + 2] += accn[j][2]; lacc[base + 3] += accn[j][3];
        if (pc == 0) { lacc[o * 18 + 16] += accr[j]; lacc[o * 18 + 17] += accd[j]; }
      }
    }
    __syncthreads();
  }

  float* pb = part + ((size_t)ch * B_ + b) * SLOT_;
  for (int idx = tid; idx < SLOT_; idx += 256) pb[idx] = lacc[idx];
}

// -------------------------------------------------------------------------
// Iteration finalize: reduce chunk partials (fixed order), update A_h + Mu.
// Last iteration writes d_out = [Mu (B,O,P) | A_h (B,O)].
// -------------------------------------------------------------------------
__global__ __launch_bounds__(64) void routing_finalize(const float* __restrict__ part,
                                                       float* __restrict__ mu,
                                                       float* __restrict__ Ah,
                                                       float* __restrict__ logA,
                                                       float* __restrict__ out,
                                                       int last) {
  const int b = blockIdx.x, o = threadIdx.x;
  float num[16];
#pragma unroll
  for (int p = 0; p < 16; ++p) num[p] = 0.0f;
  float rs = 0.0f, den = 0.0f;
  for (int ch = 0; ch < CH_; ++ch) {
    const float* pb = part + ((size_t)ch * B_ + b) * SLOT_ + o * 18;
#pragma unroll
    for (int p = 0; p < 16; ++p) num[p] += pb[p];
    rs  += pb[16];
    den += pb[17];
  }
  __shared__ float s[64];
  s[o] = rs;
  __syncthreads();
  float tot = 0.0f;
  for (int k = 0; k < 64; ++k) tot += s[k];
  tot += EPS_;
  float A = rs / tot;
  A = fminf(fmaxf(A, EPS_), 1.0f - EPS_);
  den += EPS_;
  float q[16], sq = 0.0f;
#pragma unroll
  for (int p = 0; p < 16; ++p) { q[p] = num[p] / den; sq += q[p] * q[p]; }
  const float rn = rsqrtf(fmaxf(sq, 1e-12f));
  if (last) {
#pragma unroll
    for (int p = 0; p < 16; ++p) out[(size_t)b * 1024 + o * 16 + p] = q[p] * rn;
    out[32768 + b * 64 + o] = A;
  } else {
#pragma unroll
    for (int p = 0; p < 16; ++p) mu[(size_t)b * 1024 + o * 16 + p] = q[p] * rn;
    Ah[b * 64 + o]   = A;
    logA[b * 64 + o] = logf(A + EPS_);
  }
}

// -------------------------------------------------------------------------
extern "C" void kernel_launch(void* const* d_in, const int* in_sizes, int n_in,
                              void* d_out, int out_size, void* d_ws, size_t ws_size,
                              hipStream_t stream) {
  (void)in_sizes; (void)n_in; (void)out_size; (void)ws_size;
  const float* votes = (const float*)d_in[0];
  const float* act   = (const float*)d_in[1];
  float* out  = (float*)d_out;
  float* ws   = (float*)d_ws;
  float* part = ws + WS_PART;
  float* mu   = ws + WS_MU;
  float* Ah   = ws + WS_A;
  float* logA = ws + WS_LOGA;

  const dim3 gridS(B_ * CH_), blk(256);
  sum_votes_wmma<<<gridS, blk, 0, stream>>>(votes, part);
  init_finalize<<<dim3(B_), dim3(64), 0, stream>>>(part, mu, Ah, logA);
  for (int it = 0; it < 3; ++it) {
    routing_pass<<<gridS, blk, 0, stream>>>(votes, act, mu, Ah, logA, part);
    routing_finalize<<<dim3(B_), dim3(64), 0, stream>>>(part, mu, Ah, logA, out, it == 2);
  }
}